// SelfAttention_BigBird_24026047054596
// MI455X (gfx1250) — compile-verified
//
#include <hip/hip_runtime.h>
#include <hip/hip_bf16.h>
#include <math.h>

// Problem constants (from reference)
#define BB   2
#define LL   2048
#define CC   256
#define HH   8
#define DKK  32
#define MROWS (BB * LL)          // 4096 GEMM rows
#define SCALE 0.17677669529663687f  // 1/sqrt(32)

typedef __attribute__((ext_vector_type(2))) float v2f;
typedef __attribute__((ext_vector_type(8))) float v8f;

// ---------------------------------------------------------------------------
// WMMA f32 GEMM: out = X(M x K) @ W(K x N) + bias, M=4096, N=K=256.
// One 16x16 output tile per wave, V_WMMA_F32_16X16X4_F32, K stepped by 4.
// qkv_layout=1 stores into (B,H,L,DK); else row-major (M,N).
// ---------------------------------------------------------------------------
__global__ __launch_bounds__(256) void gemm_wmma_f32(
    const float* __restrict__ X, const float* __restrict__ W,
    const float* __restrict__ bias, float* __restrict__ out, int qkv_layout)
{
    constexpr int N = 256, K = 256;
    const int lane   = threadIdx.x & 31;
    const int wave   = threadIdx.x >> 5;
    const int tile   = blockIdx.x * 8 + wave;       // 4096 tiles total
    const int tile_m = tile >> 4;                   // N/16 == 16 tiles per row
    const int tile_n = tile & 15;

    const int half = lane >> 4;                     // 0: lanes 0-15, 1: 16-31
    const int mn   = lane & 15;

    v8f c = {0.f, 0.f, 0.f, 0.f, 0.f, 0.f, 0.f, 0.f};

    const float* xrow = X + (size_t)(tile_m * 16 + mn) * K;   // A row for this lane
    const float* wcol = W + tile_n * 16 + mn;                 // B column for this lane

    #pragma unroll 8
    for (int k = 0; k < K; k += 4) {
        const int kb = k + half * 2;
        v2f a, b;
        a.x = xrow[kb];                 // A: (m, kb), (m, kb+1)  -> VGPR0/1 per ISA layout
        a.y = xrow[kb + 1];
        b.x = wcol[(size_t)kb * N];     // B: (kb, n), (kb+1, n)
        b.y = wcol[(size_t)(kb + 1) * N];
        c = __builtin_amdgcn_wmma_f32_16x16x4_f32(
                /*neg_a=*/false, a, /*neg_b=*/false, b,
                /*c_mod=*/(short)0, c, /*reuse_a=*/false, /*reuse_b=*/false);
    }

    const int col = tile_n * 16 + mn;
    const float bv = bias[col];

    #pragma unroll
    for (int i = 0; i < 8; ++i) {
        const int row = tile_m * 16 + i + half * 8;   // C/D layout: M = i + 8*(lane>=16)
        const float val = c[i] + bv;
        if (qkv_layout) {
            const int b_ = row >> 11, l_ = row & (LL - 1);
            const int h_ = col >> 5,  d_ = col & 31;
            out[(((size_t)(b_ * HH + h_) * LL + l_) << 5) + d_] = val;
        } else {
            out[(size_t)row * N + col] = val;
        }
    }
}

// ---------------------------------------------------------------------------
// Per-(b,h) column sum of V: Vsum[bh][d] = sum_l V[bh][l][d]
// ---------------------------------------------------------------------------
__global__ __launch_bounds__(256) void vsum_kernel(const float* __restrict__ V,
                                                   float* __restrict__ Vsum)
{
    const int bh = blockIdx.x;              // 16 blocks
    const int d = threadIdx.x & 31;
    const int chunk = threadIdx.x >> 5;     // 8 chunks of 256 rows
    float acc = 0.f;
    const float* base = V + ((size_t)bh * LL) * DKK;
    for (int l = chunk * 256; l < chunk * 256 + 256; ++l)
        acc += base[(size_t)l * DKK + d];
    __shared__ float s[8][32];
    s[chunk][d] = acc;
    __syncthreads();
    if (chunk == 0) {
        float t = 0.f;
        #pragma unroll
        for (int w = 0; w < 8; ++w) t += s[w][d];
        Vsum[bh * DKK + d] = t;
    }
}

__device__ __forceinline__ float wave_sum(float v) {
    #pragma unroll
    for (int off = 16; off > 0; off >>= 1) v += __shfl_xor(v, off, 32);
    return v;
}
__device__ __forceinline__ float wave_max(float v) {
    #pragma unroll
    for (int off = 16; off > 0; off >>= 1) v = fmaxf(v, __shfl_xor(v, off, 32));
    return v;
}

// ---------------------------------------------------------------------------
// Band rows l in [1, L-2]: closed-form softmax over {0, l-1, l, l+1, L-1}
// plus the (L - |S|) zero-score positions. One wave per (b,h,l); lane = dim.
// ---------------------------------------------------------------------------
__global__ __launch_bounds__(256) void band_rows_kernel(
    const float* __restrict__ Q, const float* __restrict__ K,
    const float* __restrict__ V, const float* __restrict__ Vsum,
    float* __restrict__ Zout)
{
    const int wid  = (blockIdx.x * blockDim.x + threadIdx.x) >> 5; // wave id
    const int lane = threadIdx.x & 31;
    const int bh = wid / (LL - 2);
    const int l  = 1 + wid % (LL - 2);

    const size_t bhbase = (size_t)bh * LL;
    const float qd = Q[(bhbase + l) * DKK + lane];

    int ms[5];
    int cnt = 0;
    ms[cnt++] = 0;
    if (l - 1 != 0)      ms[cnt++] = l - 1;
    ms[cnt++] = l;
    if (l + 1 != LL - 1) ms[cnt++] = l + 1;
    ms[cnt++] = LL - 1;

    float s[5];
    float smax = 0.f;                       // zero-class score is 0
    for (int i = 0; i < cnt; ++i) {
        float p = qd * K[(bhbase + ms[i]) * DKK + lane];
        p = wave_sum(p) * SCALE;
        s[i] = p;
        smax = fmaxf(smax, p);
    }

    const float t = __expf(-smax);          // weight of each zero-score column
    float den = t * (float)(LL - cnt);
    float acc = 0.f;
    for (int i = 0; i < cnt; ++i) {
        const float e = __expf(s[i] - smax);
        den += e;
        acc += (e - t) * V[(bhbase + ms[i]) * DKK + lane];
    }

    const float z = (t * Vsum[bh * DKK + lane] + acc) / den;
    const int b_ = bh >> 3, h_ = bh & 7;
    Zout[((size_t)b_ * LL + l) * CC + h_ * DKK + lane] = z;
}

// ---------------------------------------------------------------------------
// Global rows (l = 0 and l = L-1): fully dense softmax(q.K)V over 2048 keys.
// One 256-thread block per (b,h,g).
// ---------------------------------------------------------------------------
__global__ __launch_bounds__(256) void global_rows_kernel(
    const float* __restrict__ Q, const float* __restrict__ K,
    const float* __restrict__ V, float* __restrict__ Zout)
{
    __shared__ float s_sc[LL];
    __shared__ float s_q[DKK];
    __shared__ float s_red[8];
    __shared__ float s_acc[8][32];

    const int bh = blockIdx.x >> 1;
    const int g  = (blockIdx.x & 1) ? (LL - 1) : 0;
    const int tid = threadIdx.x;
    const int wave = tid >> 5, lane = tid & 31;
    const size_t bhbase = (size_t)bh * LL;

    if (tid < DKK) s_q[tid] = Q[(bhbase + g) * DKK + tid];
    __syncthreads();

    // scores
    float lmax = -3.4e38f;
    for (int j = 0; j < 8; ++j) {
        const int m = tid + j * 256;
        const float* kr = K + (bhbase + m) * DKK;
        float dot = 0.f;
        #pragma unroll
        for (int d = 0; d < DKK; ++d) dot += s_q[d] * kr[d];
        dot *= SCALE;
        s_sc[m] = dot;
        lmax = fmaxf(lmax, dot);
    }
    lmax = wave_max(lmax);
    if (lane == 0) s_red[wave] = lmax;
    __syncthreads();
    float gmax = s_red[0];
    #pragma unroll
    for (int w = 1; w < 8; ++w) gmax = fmaxf(gmax, s_red[w]);
    __syncthreads();

    // exp + denominator
    float lsum = 0.f;
    for (int j = 0; j < 8; ++j) {
        const int m = tid + j * 256;
        const float e = __expf(s_sc[m] - gmax);
        s_sc[m] = e;
        lsum += e;
    }
    lsum = wave_sum(lsum);
    if (lane == 0) s_red[wave] = lsum;
    __syncthreads();
    float den = 0.f;
    #pragma unroll
    for (int w = 0; w < 8; ++w) den += s_red[w];

    // weighted sum of V: wave 'w' handles keys [w*256, w*256+256); lane = dim
    float acc = 0.f;
    const float* vbase = V + bhbase * DKK;
    for (int m = wave * 256; m < wave * 256 + 256; ++m)
        acc += s_sc[m] * vbase[(size_t)m * DKK + lane];
    s_acc[wave][lane] = acc;
    __syncthreads();

    if (wave == 0) {
        float z = 0.f;
        #pragma unroll
        for (int w = 0; w < 8; ++w) z += s_acc[w][lane];
        z /= den;
        const int b_ = bh >> 3, h_ = bh & 7;
        Zout[((size_t)b_ * LL + g) * CC + h_ * DKK + lane] = z;
    }
}

// ---------------------------------------------------------------------------
extern "C" void kernel_launch(void* const* d_in, const int* in_sizes, int n_in,
                              void* d_out, int out_size, void* d_ws, size_t ws_size,
                              hipStream_t stream) {
    (void)in_sizes; (void)n_in; (void)out_size; (void)ws_size;

    const float* qx = (const float*)d_in[0];
    const float* kx = (const float*)d_in[1];
    const float* vx = (const float*)d_in[2];
    const float* WQ = (const float*)d_in[3];
    const float* bQ = (const float*)d_in[4];
    const float* WK = (const float*)d_in[5];
    const float* bK = (const float*)d_in[6];
    const float* WV = (const float*)d_in[7];
    const float* bV = (const float*)d_in[8];
    const float* WO = (const float*)d_in[9];
    const float* bO = (const float*)d_in[10];
    float* out = (float*)d_out;

    const size_t QKV_ELEMS = (size_t)BB * HH * LL * DKK;   // 1,048,576
    const size_t Z_ELEMS   = (size_t)BB * LL * CC;         // 1,048,576
    float* Qw   = (float*)d_ws;
    float* Kw   = Qw + QKV_ELEMS;
    float* Vw   = Kw + QKV_ELEMS;
    float* Zw   = Vw + QKV_ELEMS;
    float* Vsum = Zw + Z_ELEMS;

    const dim3 gemm_grid(512), blk(256);   // 4096 tiles, 8 waves/block

    // Projections (fused bias + reshape to (B,H,L,DK))
    gemm_wmma_f32<<<gemm_grid, blk, 0, stream>>>(qx, WQ, bQ, Qw, 1);
    gemm_wmma_f32<<<gemm_grid, blk, 0, stream>>>(kx, WK, bK, Kw, 1);
    gemm_wmma_f32<<<gemm_grid, blk, 0, stream>>>(vx, WV, bV, Vw, 1);

    // Per-(b,h) V column sums for the zero-score columns of softmax
    vsum_kernel<<<dim3(BB * HH), blk, 0, stream>>>(Vw, Vsum);

    // Band rows: 16 * 2046 waves = 32736 -> exactly 4092 blocks of 8 waves
    band_rows_kernel<<<dim3(4092), blk, 0, stream>>>(Qw, Kw, Vw, Vsum, Zw);

    // Global rows 0 and L-1: one block per (b,h,g)
    global_rows_kernel<<<dim3(BB * HH * 2), blk, 0, stream>>>(Qw, Kw, Vw, Zw);

    // Output projection -> d_out (B, L, C)
    gemm_wmma_f32<<<gemm_grid, blk, 0, stream>>>(Zw, WO, bO, out, 0);
}